// GraphDegreeConv_63874753626413
// MI455X (gfx1250) — compile-verified
//
#include <hip/hip_runtime.h>
#include <hip/hip_bf16.h>

typedef float v2f __attribute__((ext_vector_type(2)));
typedef float v8f __attribute__((ext_vector_type(8)));

#ifndef __has_builtin
#define __has_builtin(x) 0
#endif
#if __has_builtin(__builtin_amdgcn_wmma_f32_16x16x4_f32)
#define HAVE_WMMA_F32X4 1
#else
#define HAVE_WMMA_F32X4 0
#endif

constexpr int kNodes   = 100000;
constexpr int kPerDeg  = 25000;
constexpr int kNodeF   = 256;
constexpr int kEdgeF   = 128;
constexpr int kOut     = 256;
constexpr int ROWS     = 64;            // rows of output per block
constexpr int XS       = 644;           // LDS row stride (floats): 16B-aligned, bank-conflict-free
constexpr int THREADS  = 512;           // 16 waves; wave w owns output cols [16w, 16w+16)
constexpr int BLK_PER_DEG = (kPerDeg + ROWS - 1) / ROWS;   // 391
constexpr int TOT_BLKS    = BLK_PER_DEG * 4;               // 1564
constexpr float BN_EPS = 1e-5f;
constexpr size_t SMEM_BYTES = (size_t)ROWS * XS * sizeof(float);  // ~161 KB (of 320 KB/WGP)

__device__ __forceinline__ v8f wmma_f32_16x16x4(v2f a, v2f b, v8f c) {
#if HAVE_WMMA_F32X4
  return __builtin_amdgcn_wmma_f32_16x16x4_f32(false, a, false, b, (short)0, c,
                                               false, false);
#else
  c[0] += a.x * b.x + a.y * b.y;   // compile-safety fallback only
  return c;
#endif
}

__global__ __launch_bounds__(THREADS, 1)
void gdc_fused_gemm(const float* __restrict__ node_repr,
                    const float* __restrict__ edge_repr,
                    const int* __restrict__ ni1, const int* __restrict__ ni2,
                    const int* __restrict__ ni3, const int* __restrict__ ni4,
                    const int* __restrict__ ei1, const int* __restrict__ ei2,
                    const int* __restrict__ ei3, const int* __restrict__ ei4,
                    const float* __restrict__ W_deg,   // [4][384][256]
                    const float* __restrict__ W_self,  // [256][256]
                    float* __restrict__ act,           // d_out staging [100000][256]
                    float* __restrict__ partials)      // [TOT_BLKS][512] (sum|sumsq)
{
  extern __shared__ float Xs[];   // [ROWS][XS]: cols 0..255 self, 256..511 node-sum, 512..639 edge-sum
  __shared__ int nIdxS[ROWS * 4];
  __shared__ int eIdxS[ROWS * 4];

  const int g    = blockIdx.y;          // degree group 0..3 (degree = g+1)
  const int deg  = g + 1;
  const int l0   = blockIdx.x * ROWS;   // local row base within degree group
  const int valid = (kPerDeg - l0 < ROWS) ? (kPerDeg - l0) : ROWS;
  const int* __restrict__ nI = (g == 0) ? ni1 : (g == 1) ? ni2 : (g == 2) ? ni3 : ni4;
  const int* __restrict__ eI = (g == 0) ? ei1 : (g == 1) ? ei2 : (g == 2) ? ei3 : ei4;
  const int tid = threadIdx.x;

  // ---------------- stage neighbor indices into LDS (one read per index) ----------------
  if (tid < ROWS * deg) {
    int l = l0 + tid / deg; if (l > kPerDeg - 1) l = kPerDeg - 1;
    const int j = tid % deg;
    nIdxS[(tid / deg) * deg + j] = nI[l * deg + j];
    eIdxS[(tid / deg) * deg + j] = eI[l * deg + j];
  }
  __syncthreads();

  // ---------------- gather phase (coalesced float4, summed over neighbors) ----------------
  for (int t = tid; t < ROWS * 64; t += THREADS) {       // self + node-sum: 64 rows x 64 float4
    const int r  = t >> 6;
    const int c4 = (t & 63) * 4;
    int l = l0 + r; if (l > kPerDeg - 1) l = kPerDeg - 1;   // clamp tail rows (masked at store)
    const float4 s = *(const float4*)(node_repr + (size_t)(g * kPerDeg + l) * kNodeF + c4);
    *(float4*)(Xs + r * XS + c4) = s;
    float4 acc = make_float4(0.f, 0.f, 0.f, 0.f);
    for (int j = 0; j < deg; ++j) {
      const int ni = nIdxS[r * deg + j];
      const float4 v = *(const float4*)(node_repr + (size_t)ni * kNodeF + c4);
      acc.x += v.x; acc.y += v.y; acc.z += v.z; acc.w += v.w;
    }
    *(float4*)(Xs + r * XS + kNodeF + c4) = acc;
  }
  for (int t = tid; t < ROWS * 32; t += THREADS) {       // edge-sum: 64 rows x 32 float4
    const int r  = t >> 5;
    const int c4 = (t & 31) * 4;
    float4 acc = make_float4(0.f, 0.f, 0.f, 0.f);
    for (int j = 0; j < deg; ++j) {
      const int ei = eIdxS[r * deg + j];
      const float4 v = *(const float4*)(edge_repr + (size_t)ei * kEdgeF + c4);
      acc.x += v.x; acc.y += v.y; acc.z += v.z; acc.w += v.w;
    }
    *(float4*)(Xs + r * XS + 2 * kNodeF + c4) = acc;
  }
  __syncthreads();

  // ---------------- WMMA phase: wave computes 64x16 tile, K = 640 ----------------
  const int wave = tid >> 5;
  const int lane = tid & 31;
  const int mlo  = lane & 15;
  const int half = lane >> 4;     // A/B: lanes 0-15 hold K=k,k+1; lanes 16-31 hold K=k+2,k+3
  const int koff = half * 2;
  const int nb   = wave * 16;     // column base of this wave's tile

  v8f c[4];
#pragma unroll
  for (int mt = 0; mt < 4; ++mt) c[mt] = (v8f)0.0f;

  const float* __restrict__ Wd = W_deg + (size_t)g * 384 * kOut;

#pragma unroll 1
  for (int seg = 0; seg < 2; ++seg) {
    const float* __restrict__ Wb = seg ? Wd : W_self;  // combined 640x256 weight, row-major
    const int klen = seg ? 384 : 256;
    const int xoff = seg ? 256 : 0;
#pragma unroll 4
    for (int k = 0; k < klen; k += 4) {
      // B fragment (4x16): this lane: column nb+mlo, rows k+koff, k+koff+1
      const float* colp = Wb + (size_t)(k + koff) * kOut + nb + mlo;
      v2f b; b.x = colp[0]; b.y = colp[kOut];
      // A fragments (16x4) from LDS, one per M-tile (b64 each, conflict-free)
      const float* ap = Xs + mlo * XS + xoff + k + koff;
      const v2f a0 = *(const v2f*)(ap + 0 * 16 * XS);
      const v2f a1 = *(const v2f*)(ap + 1 * 16 * XS);
      const v2f a2 = *(const v2f*)(ap + 2 * 16 * XS);
      const v2f a3 = *(const v2f*)(ap + 3 * 16 * XS);
      c[0] = wmma_f32_16x16x4(a0, b, c[0]);
      c[1] = wmma_f32_16x16x4(a1, b, c[1]);
      c[2] = wmma_f32_16x16x4(a2, b, c[2]);
      c[3] = wmma_f32_16x16x4(a3, b, c[3]);
    }
  }

  // ---------------- epilogue: store activations + deterministic BN partials ----------------
  // C/D layout: VGPR v -> row M = v + 8*half, col N = mlo (within 16x16 tile)
  float* __restrict__ orow = act + ((size_t)(g * kPerDeg + l0) + half * 8) * kOut + nb + mlo;
  float ps = 0.f, pq = 0.f;
  if (valid == ROWS) {
    // fast path (390/391 blocks per degree): no per-element masking, no EXEC churn
#pragma unroll
    for (int mt = 0; mt < 4; ++mt) {
#pragma unroll
      for (int v = 0; v < 8; ++v) {
        const float x = c[mt][v];
        orow[(size_t)(mt * 16 + v) * kOut] = x;
        ps += x;
        pq += x * x;
      }
    }
  } else {
#pragma unroll
    for (int mt = 0; mt < 4; ++mt) {
#pragma unroll
      for (int v = 0; v < 8; ++v) {
        const int rl = mt * 16 + half * 8 + v;
        const float x = c[mt][v];
        if (rl < valid) {
          orow[(size_t)(mt * 16 + v) * kOut] = x;
          ps += x;
          pq += x * x;
        }
      }
    }
  }
  __syncthreads();                  // all waves done reading Xs -> reuse as scratch
  float* Psum = Xs;                 // [2][256]
  float* Psq  = Xs + 512;           // [2][256]
  Psum[half * 256 + nb + mlo] = ps;
  Psq [half * 256 + nb + mlo] = pq;
  __syncthreads();
  if (tid < 256) {                  // fixed-order combine -> deterministic
    const int col = tid;
    const float s = Psum[col] + Psum[256 + col];
    const float q = Psq[col]  + Psq[256 + col];
    const size_t bl = (size_t)(g * BLK_PER_DEG + blockIdx.x) * 512;
    partials[bl + col]       = s;
    partials[bl + 256 + col] = q;
  }
}

__global__ __launch_bounds__(256)
void gdc_stats(const float* __restrict__ partials, float* __restrict__ stats) {
  const int col = threadIdx.x;      // 256 threads, 1 block: fixed-order reduction
  float s = 0.f, q = 0.f;
#pragma unroll 4
  for (int b = 0; b < TOT_BLKS; ++b) {
    s += partials[(size_t)b * 512 + col];
    q += partials[(size_t)b * 512 + 256 + col];
  }
  const float invN = 1.0f / (float)kNodes;
  const float mean = s * invN;
  const float var  = q * invN - mean * mean;   // biased var, matches jnp.var
  stats[col]       = mean;
  stats[256 + col] = rsqrtf(var + BN_EPS);
}

__global__ __launch_bounds__(256)
void gdc_bn_relu(float* __restrict__ out, const float* __restrict__ stats) {
  __shared__ float sm[512];
  sm[threadIdx.x]       = stats[threadIdx.x];
  sm[256 + threadIdx.x] = stats[256 + threadIdx.x];
  __syncthreads();
  const size_t i = (size_t)blockIdx.x * blockDim.x + threadIdx.x;   // one float4 each
  if (i < (size_t)kNodes * kOut / 4) {
    float4 v = ((const float4*)out)[i];
    const int c = (int)(i & 63) * 4;
    v.x = fmaxf(0.f, (v.x - sm[c + 0]) * sm[256 + c + 0]);
    v.y = fmaxf(0.f, (v.y - sm[c + 1]) * sm[256 + c + 1]);
    v.z = fmaxf(0.f, (v.z - sm[c + 2]) * sm[256 + c + 2]);
    v.w = fmaxf(0.f, (v.w - sm[c + 3]) * sm[256 + c + 3]);
    ((float4*)out)[i] = v;
  }
}

extern "C" void kernel_launch(void* const* d_in, const int* in_sizes, int n_in,
                              void* d_out, int out_size, void* d_ws, size_t ws_size,
                              hipStream_t stream) {
  (void)in_sizes; (void)n_in; (void)out_size; (void)ws_size;
  const float* node_repr = (const float*)d_in[0];
  const float* edge_repr = (const float*)d_in[1];
  const int* ni1 = (const int*)d_in[2];
  const int* ni2 = (const int*)d_in[3];
  const int* ni3 = (const int*)d_in[4];
  const int* ni4 = (const int*)d_in[5];
  const int* ei1 = (const int*)d_in[6];
  const int* ei2 = (const int*)d_in[7];
  const int* ei3 = (const int*)d_in[8];
  const int* ei4 = (const int*)d_in[9];
  const float* W_deg  = (const float*)d_in[10];
  const float* W_self = (const float*)d_in[11];
  // d_in[12] = bias: per-column constant cancels exactly in affine-free BatchNorm -> unused.

  float* out = (float*)d_out;
  float* wsf = (float*)d_ws;
  float* partials = wsf;                                 // TOT_BLKS*512 floats (~3.2 MB)
  float* stats    = wsf + (size_t)TOT_BLKS * 512;        // mean[256], rstd[256]

  dim3 gridA(BLK_PER_DEG, 4);
  gdc_fused_gemm<<<gridA, THREADS, SMEM_BYTES, stream>>>(
      node_repr, edge_repr, ni1, ni2, ni3, ni4, ei1, ei2, ei3, ei4,
      W_deg, W_self, out, partials);

  gdc_stats<<<1, 256, 0, stream>>>(partials, stats);

  const int nvec4 = kNodes * kOut / 4;                   // 6,400,000
  gdc_bn_relu<<<(nvec4 + 255) / 256, 256, 0, stream>>>(out, stats);
}